// TPSSpatialTransformer_45990509805881
// MI455X (gfx1250) — compile-verified
//
#include <hip/hip_runtime.h>
#include <math.h>

#define OUT_H 128
#define OUT_W 512
#define NUM_CTRL 20
#define KDIM 23          // 3 + NUM_CTRL
#define KPAD 24          // padded K for 6 chunks of 4
#define EPS 1e-6f
#define NCH 3

typedef __attribute__((ext_vector_type(2))) float v2f;
typedef __attribute__((ext_vector_type(8))) float v8f;

// ---------------------------------------------------------------------------
// Kernel 1: per-batch TPS system solve (23x23, 2 RHS), Gaussian elimination
// with partial pivoting. One thread per batch (64 threads total -> negligible).
// Output: T[b][2][KPAD] (row-major, padded with zeros at k=23).
// ---------------------------------------------------------------------------
__global__ void tps_solve_kernel(const float* __restrict__ coord,  // [B,20,2]
                                 const float* __restrict__ src,    // [20,2]
                                 float* __restrict__ T,            // [B,2,KPAD]
                                 int B) {
  int b = blockIdx.x * blockDim.x + threadIdx.x;
  if (b >= B) return;

  float cx[NUM_CTRL], cy[NUM_CTRL];
  for (int k = 0; k < NUM_CTRL; ++k) {
    cx[k] = coord[(b * NUM_CTRL + k) * 2 + 0];
    cy[k] = coord[(b * NUM_CTRL + k) * 2 + 1];
  }

  float W[KDIM][KDIM];
  float R[KDIM][2];
  for (int i = 0; i < KDIM; ++i)
    for (int j = 0; j < KDIM; ++j) W[i][j] = 0.0f;

  // W0 = [P | R], P row = (1, x, y); R_ij = d2*log(d2+eps)
  for (int i = 0; i < NUM_CTRL; ++i) {
    W[i][0] = 1.0f; W[i][1] = cx[i]; W[i][2] = cy[i];
    for (int k = 0; k < NUM_CTRL; ++k) {
      float dx = cx[i] - cx[k], dy = cy[i] - cy[k];
      float d2 = dx * dx + dy * dy;
      W[i][3 + k] = d2 * logf(d2 + EPS);
    }
    R[i][0] = src[i * 2 + 0];
    R[i][1] = src[i * 2 + 1];
  }
  // W1 = [0_{3x3} | P^T]
  for (int a = 0; a < 3; ++a) {
    for (int k = 0; k < NUM_CTRL; ++k) {
      float pv = (a == 0) ? 1.0f : ((a == 1) ? cx[k] : cy[k]);
      W[NUM_CTRL + a][3 + k] = pv;
    }
    R[NUM_CTRL + a][0] = 0.0f;
    R[NUM_CTRL + a][1] = 0.0f;
  }

  // LU-style elimination with partial pivoting
  for (int col = 0; col < KDIM; ++col) {
    int piv = col;
    float best = fabsf(W[col][col]);
    for (int r = col + 1; r < KDIM; ++r) {
      float v = fabsf(W[r][col]);
      if (v > best) { best = v; piv = r; }
    }
    if (piv != col) {
      for (int j = col; j < KDIM; ++j) { float t = W[col][j]; W[col][j] = W[piv][j]; W[piv][j] = t; }
      float t0 = R[col][0]; R[col][0] = R[piv][0]; R[piv][0] = t0;
      float t1 = R[col][1]; R[col][1] = R[piv][1]; R[piv][1] = t1;
    }
    float inv = 1.0f / W[col][col];
    for (int r = col + 1; r < KDIM; ++r) {
      float f = W[r][col] * inv;
      for (int j = col + 1; j < KDIM; ++j) W[r][j] -= f * W[col][j];
      R[r][0] -= f * R[col][0];
      R[r][1] -= f * R[col][1];
    }
  }
  // Back substitution
  float X[KDIM][2];
  for (int r = KDIM - 1; r >= 0; --r) {
    float s0 = R[r][0], s1 = R[r][1];
    for (int j = r + 1; j < KDIM; ++j) { s0 -= W[r][j] * X[j][0]; s1 -= W[r][j] * X[j][1]; }
    float inv = 1.0f / W[r][r];
    X[r][0] = s0 * inv;
    X[r][1] = s1 * inv;
  }
  // Store transposed (2 x KPAD), zero-padded
  float* Tb = T + (size_t)b * 2 * KPAD;
  for (int j = 0; j < KPAD; ++j) {
    Tb[j]        = (j < KDIM) ? X[j][0] : 0.0f;
    Tb[KPAD + j] = (j < KDIM) ? X[j][1] : 0.0f;
  }
}

// ---------------------------------------------------------------------------
// Kernel 2: fused grid + (T x grid) via V_WMMA_F32_16X16X4_F32 + bilinear
// gather. One wave = 32 pixels = two 16-pixel N-tiles, 6 chained K=4 WMMAs
// per tile (K = 24 padded). fp32 end to end.
//
// Fragment layouts (ISA 7.12.2, f32):
//   A 16x4 : lane m=lane&15, half=lane>>4; VGPR v -> K = v + 2*half
//   B 4x16 : lane n=lane&15, half;         VGPR v -> K row = v + 2*half
//   D 16x16: lane n=lane&15, half;         VGPR v -> M row = v + 8*half
// ---------------------------------------------------------------------------
__global__ void __launch_bounds__(256)
tps_warp_kernel(const float* __restrict__ img,    // [B,3,128,512]
                const float* __restrict__ coord,  // [B,20,2]
                const float* __restrict__ T,      // [B,2,KPAD]
                float* __restrict__ out) {        // [B,3,128,512]
  __shared__ float s_cx[NUM_CTRL];
  __shared__ float s_cy[NUM_CTRL];
  __shared__ float s_T[2 * KPAD];

  const int HW = OUT_H * OUT_W;            // 65536
  const int blocksPerBatch = HW / 256;     // 256
  const int b = blockIdx.x / blocksPerBatch;
  const int chunk = blockIdx.x % blocksPerBatch;
  const int tid = threadIdx.x;

  if (tid < NUM_CTRL) {
    s_cx[tid] = coord[(b * NUM_CTRL + tid) * 2 + 0];
    s_cy[tid] = coord[(b * NUM_CTRL + tid) * 2 + 1];
  }
  if (tid < 2 * KPAD) s_T[tid] = T[(size_t)b * 2 * KPAD + tid];
  __syncthreads();

  const int lane = tid & 31;
  const int wave = tid >> 5;
  const int pbase = chunk * 256 + wave * 32;  // first of this wave's 32 pixels
  const int n = lane & 15;
  const int half = lane >> 4;

  // Normalized output-grid coords for the two sub-tile pixels this lane covers
  const int p0 = pbase + n;
  const int p1 = pbase + 16 + n;
  const float x0p = -1.0f + (float)(p0 & (OUT_W - 1)) * (2.0f / (float)(OUT_W - 1));
  const float y0p = -1.0f + (float)(p0 >> 9) * (2.0f / (float)(OUT_H - 1));
  const float x1p = -1.0f + (float)(p1 & (OUT_W - 1)) * (2.0f / (float)(OUT_W - 1));
  const float y1p = -1.0f + (float)(p1 >> 9) * (2.0f / (float)(OUT_H - 1));

  // grid row value: row0=1, row1=x, row2=y, row 3+k = d2*log(d2+eps), row23=0
  auto gridval = [&](int row, float x, float y) -> float {
    int k = row - 3;
    int kc = (k < 0) ? 0 : ((k > NUM_CTRL - 1) ? NUM_CTRL - 1 : k);
    float dx = x - s_cx[kc];
    float dy = y - s_cy[kc];
    float d2 = dx * dx + dy * dy;
    float v = d2 * logf(d2 + EPS);
    v = (row == 0) ? 1.0f : v;
    v = (row == 1) ? x : v;
    v = (row == 2) ? y : v;
    v = (row >= KDIM) ? 0.0f : v;
    return v;
  };

  v8f c0 = {};  // D for pixels pbase..pbase+15   (rows 0,1 = x_s,y_s)
  v8f c1 = {};  // D for pixels pbase+16..pbase+31
#pragma unroll
  for (int t = 0; t < KPAD / 4; ++t) {
    const int r0 = 4 * t + 2 * half;
    const int r1 = r0 + 1;
    // A fragment: rows of T (M=0 -> x_s coeffs, M=1 -> y_s coeffs, else 0)
    const int ms = n & 1;
    const float am = (n < 2) ? 1.0f : 0.0f;
    v2f a;
    a[0] = am * s_T[ms * KPAD + r0];
    a[1] = am * s_T[ms * KPAD + r1];
    // B fragments: grid values for this lane's (K rows, pixel column)
    v2f bb0, bb1;
    bb0[0] = gridval(r0, x0p, y0p);
    bb0[1] = gridval(r1, x0p, y0p);
    bb1[0] = gridval(r0, x1p, y1p);
    bb1[1] = gridval(r1, x1p, y1p);
    c0 = __builtin_amdgcn_wmma_f32_16x16x4_f32(false, a, false, bb0, (short)0, c0, false, false);
    c1 = __builtin_amdgcn_wmma_f32_16x16x4_f32(false, a, false, bb1, (short)0, c1, false, false);
  }

  // Lanes 0-15 hold tile0 (x_s,y_s) in c0[0],c0[1]; tile1's live in lanes
  // 0-15 of c1 -> shuffle them over to lanes 16-31.
  float xs_t1 = __shfl(c1[0], lane & 15, 32);
  float ys_t1 = __shfl(c1[1], lane & 15, 32);
  float xs = (lane < 16) ? c0[0] : xs_t1;
  float ys = (lane < 16) ? c0[1] : ys_t1;

  // Bilinear sampling (exact reference semantics: clip indices, weights from
  // clipped index floats).
  const int p = pbase + lane;
  float X = (xs + 1.0f) * ((float)OUT_W * 0.5f);
  float Y = (ys + 1.0f) * ((float)OUT_H * 0.5f);
  float fx0 = floorf(X);
  float fy0 = floorf(Y);
  float x0cf = fminf(fmaxf(fx0, 0.0f), (float)(OUT_W - 1));
  float x1cf = fminf(fmaxf(fx0 + 1.0f, 0.0f), (float)(OUT_W - 1));
  float y0cf = fminf(fmaxf(fy0, 0.0f), (float)(OUT_H - 1));
  float y1cf = fminf(fmaxf(fy0 + 1.0f, 0.0f), (float)(OUT_H - 1));
  int x0c = (int)x0cf, x1c = (int)x1cf, y0c = (int)y0cf, y1c = (int)y1cf;
  float wa = (x1cf - X) * (y1cf - Y);
  float wb = (x1cf - X) * (Y - y0cf);
  float wc = (X - x0cf) * (y1cf - Y);
  float wd = (X - x0cf) * (Y - y0cf);

  const float* imgb = img + (size_t)b * NCH * HW;
  float* outb = out + (size_t)b * NCH * HW;
#pragma unroll
  for (int ch = 0; ch < NCH; ++ch) {
    const float* pc = imgb + ch * HW;
    float Ia = pc[y0c * OUT_W + x0c];
    float Ib = pc[y1c * OUT_W + x0c];
    float Ic = pc[y0c * OUT_W + x1c];
    float Id = pc[y1c * OUT_W + x1c];
    outb[ch * HW + p] = wa * Ia + wb * Ib + wc * Ic + wd * Id;
  }
}

extern "C" void kernel_launch(void* const* d_in, const int* in_sizes, int n_in,
                              void* d_out, int out_size, void* d_ws, size_t ws_size,
                              hipStream_t stream) {
  const float* img   = (const float*)d_in[0];  // [B,3,128,512]
  const float* coord = (const float*)d_in[1];  // [B,20,2]
  const float* src   = (const float*)d_in[2];  // [20,2]
  float* out = (float*)d_out;
  float* T   = (float*)d_ws;                   // [B,2,KPAD] scratch

  const int B = in_sizes[1] / (NUM_CTRL * 2);  // 64

  tps_solve_kernel<<<(B + 63) / 64, 64, 0, stream>>>(coord, src, T, B);

  const int HW = OUT_H * OUT_W;
  const int blocksPerBatch = HW / 256;
  tps_warp_kernel<<<B * blocksPerBatch, 256, 0, stream>>>(img, coord, T, out);
}